// StockGAT_63814624084274
// MI455X (gfx1250) — compile-verified
//
#include <hip/hip_runtime.h>
#include <hip/hip_bf16.h>

#define HEADS 4
#define HC    16
#define DIM   64
#define NEGS  0.2f

typedef float v2f __attribute__((ext_vector_type(2)));
typedef float v8f __attribute__((ext_vector_type(8)));

// ---------- monotonic float <-> uint key for atomic max over floats ----------
__device__ __forceinline__ unsigned f2key(float f) {
    unsigned u = __float_as_uint(f);
    return u ^ ((unsigned)(((int)u) >> 31) | 0x80000000u);
}
__device__ __forceinline__ float key2f(unsigned k) {
    unsigned u = (k & 0x80000000u) ? (k ^ 0x80000000u) : ~k;
    return __uint_as_float(u);
}

// ---------- generic zero ----------
__global__ void k_zero(float* __restrict__ p, long n) {
    long i = (long)blockIdx.x * blockDim.x + threadIdx.x;
    long st = (long)gridDim.x * blockDim.x;
    for (; i < n; i += st) p[i] = 0.0f;
}

// ---------- self-loop attr: segment count + sum over dst ----------
__global__ void k_edge_count(const int* __restrict__ ei, const float* __restrict__ eattr,
                             float* __restrict__ lsum, float* __restrict__ cnt, int E) {
    int e = blockIdx.x * blockDim.x + threadIdx.x;
    if (e >= E) return;
    int d = ei[E + e];
    atomicAdd(&lsum[d], eattr[e]);
    atomicAdd(&cnt[d], 1.0f);
}
__global__ void k_loop_attr(float* __restrict__ lsum, const float* __restrict__ cnt, int N) {
    int n = blockIdx.x * blockDim.x + threadIdx.x;
    if (n >= N) return;
    lsum[n] = lsum[n] / fmaxf(cnt[n], 1.0f);
}

// ---------- per-head edge scalar: s[h] = sum_c We[h*16+c]*att_e[h*16+c] ----------
__global__ void k_edge_scal(const float* __restrict__ We1, const float* __restrict__ ae1,
                            const float* __restrict__ We2, const float* __restrict__ ae2,
                            float* __restrict__ sed) {
    int t = threadIdx.x;       // 0..7
    if (t >= 8) return;
    const float* We = (t < 4) ? We1 : We2;
    const float* ae = (t < 4) ? ae1 : ae2;
    int h = t & 3;
    float s = 0.0f;
    for (int c = 0; c < HC; ++c) s += We[h * HC + c] * ae[h * HC + c];
    sed[t] = s;
}

// ---------- WMMA f32 GEMM: Y[N,64] = X[N,64] @ W[64,64] ----------
// 8 waves / block -> 32 rows x 64 cols per block
__global__ void k_gemm64_wmma(const float* __restrict__ X, const float* __restrict__ W,
                              float* __restrict__ Y, int nrows) {
    int wave = threadIdx.x >> 5;           // 0..7
    int lane = threadIdx.x & 31;
    int rowTile = blockIdx.x * 2 + (wave >> 2);
    int row0 = rowTile * 16;
    int col0 = (wave & 3) * 16;
    if (row0 >= nrows) return;
    int lr = lane & 15;                    // 0..15
    int hi = lane >> 4;                    // 0 or 1

    int arow = row0 + lr; if (arow > nrows - 1) arow = nrows - 1;   // clamp (N%16==0 normally)
    const float* ap = X + (size_t)arow * DIM;

    v8f acc = {};
    for (int k = 0; k < DIM; k += 4) {
        // A 16x4: lane holds A[lr][k + 2*hi + r], r=0,1
        v2f a;
        a.x = ap[k + 2 * hi + 0];
        a.y = ap[k + 2 * hi + 1];
        // B 4x16: VGPR v holds W[k + v + 2*hi][col0 + lr]
        const float* bp = W + (size_t)(k + 2 * hi) * DIM + col0 + lr;
        v2f b;
        b.x = bp[0];
        b.y = bp[DIM];
        acc = __builtin_amdgcn_wmma_f32_16x16x4_f32(false, a, false, b,
                                                    (short)0, acc, false, false);
    }
    // D 16x16: VGPR r -> row (r + 8*hi), col lr
    for (int r = 0; r < 8; ++r) {
        int row = row0 + r + 8 * hi;
        if (row < nrows) Y[(size_t)row * DIM + col0 + lr] = acc[r];
    }
}

// ---------- attention coefficients: a_src[n,h], a_dst[n,h] ----------
__global__ void k_attn_coef(const float* __restrict__ xs, const float* __restrict__ as,
                            const float* __restrict__ ad, float* __restrict__ a_src,
                            float* __restrict__ a_dst, int N) {
    int t = blockIdx.x * blockDim.x + threadIdx.x;
    if (t >= N * HEADS) return;
    int n = t >> 2, h = t & 3;
    const float* xp = xs + (size_t)n * DIM + h * HC;
    float ss = 0.0f, sd = 0.0f;
    for (int c = 0; c < HC; ++c) {
        float x = xp[c];
        ss += x * as[h * HC + c];
        sd += x * ad[h * HC + c];
    }
    a_src[t] = ss;
    a_dst[t] = sd;
}

// ---------- edge pass 1: atomic-max of leaky-relu logits per (dst, head) ----------
__global__ void k_edge_max(const int* __restrict__ ei, const float* __restrict__ eattr,
                           const float* __restrict__ lattr, const float* __restrict__ sed,
                           const float* __restrict__ a_src, const float* __restrict__ a_dst,
                           unsigned* __restrict__ mkey, int E, int N) {
    int e = blockIdx.x * blockDim.x + threadIdx.x;
    int ET = E + N;
    if (e >= ET) return;
    int s, d; float ea;
    if (e < E) { s = ei[e]; d = ei[E + e]; ea = eattr[e]; }
    else       { s = d = e - E;            ea = lattr[s]; }
    #pragma unroll
    for (int h = 0; h < HEADS; ++h) {
        float al = a_src[s * HEADS + h] + a_dst[d * HEADS + h] + ea * sed[h];
        al = (al > 0.0f) ? al : NEGS * al;
        atomicMax(&mkey[d * HEADS + h], f2key(al));
    }
}

// ---------- edge pass 2: ex = exp(alpha - max); atomic-sum denominators ----------
__global__ void k_edge_exp(const int* __restrict__ ei, const float* __restrict__ eattr,
                           const float* __restrict__ lattr, const float* __restrict__ sed,
                           const float* __restrict__ a_src, const float* __restrict__ a_dst,
                           const unsigned* __restrict__ mkey, float* __restrict__ exv,
                           float* __restrict__ den, int E, int N) {
    int e = blockIdx.x * blockDim.x + threadIdx.x;
    int ET = E + N;
    if (e >= ET) return;
    int s, d; float ea;
    if (e < E) { s = ei[e]; d = ei[E + e]; ea = eattr[e]; }
    else       { s = d = e - E;            ea = lattr[s]; }
    #pragma unroll
    for (int h = 0; h < HEADS; ++h) {
        float al = a_src[s * HEADS + h] + a_dst[d * HEADS + h] + ea * sed[h];
        al = (al > 0.0f) ? al : NEGS * al;
        float m = key2f(mkey[d * HEADS + h]);
        float ex = expf(al - m);
        exv[(size_t)e * HEADS + h] = ex;
        atomicAdd(&den[d * HEADS + h], ex);
    }
}

// ---------- edge pass 3: weighted message aggregation (one thread per edge-feature) ----------
__global__ void k_edge_agg(const int* __restrict__ ei, const float* __restrict__ exv,
                           const float* __restrict__ den, const float* __restrict__ xs,
                           float* __restrict__ agg, int E, int N) {
    long t = (long)blockIdx.x * blockDim.x + threadIdx.x;
    long ET = (long)E + N;
    if (t >= ET * DIM) return;
    long e = t >> 6;
    int f = (int)(t & 63);
    int s, d;
    if (e < E) { s = ei[e]; d = ei[E + e]; }
    else       { s = d = (int)(e - E); }
    int h = f >> 4;
    float w = exv[e * HEADS + h] / (den[d * HEADS + h] + 1e-16f);
    atomicAdd(&agg[(size_t)d * DIM + f], xs[(size_t)s * DIM + f] * w);
}

// ---------- post: out = elu(layernorm(agg + b + residual)); wave32 per node ----------
__global__ void k_post(const float* __restrict__ agg, const float* __restrict__ res,
                       const float* __restrict__ b, const float* __restrict__ g,
                       const float* __restrict__ beta, float* __restrict__ out, int N) {
    int wave = threadIdx.x >> 5;
    int lane = threadIdx.x & 31;
    int n = blockIdx.x * 8 + wave;
    if (n >= N) return;
    const float* ap = agg + (size_t)n * DIM;
    const float* rp = res + (size_t)n * DIM;
    float v0 = ap[lane]      + b[lane]      + rp[lane];
    float v1 = ap[lane + 32] + b[lane + 32] + rp[lane + 32];
    float sum = v0 + v1;
    #pragma unroll
    for (int m = 16; m >= 1; m >>= 1) sum += __shfl_xor(sum, m, 32);
    float mu = sum * (1.0f / DIM);
    float d0 = v0 - mu, d1 = v1 - mu;
    float vs = d0 * d0 + d1 * d1;
    #pragma unroll
    for (int m = 16; m >= 1; m >>= 1) vs += __shfl_xor(vs, m, 32);
    float rstd = rsqrtf(vs * (1.0f / DIM) + 1e-5f);
    float y0 = d0 * rstd * g[lane]      + beta[lane];
    float y1 = d1 * rstd * g[lane + 32] + beta[lane + 32];
    y0 = (y0 > 0.0f) ? y0 : (expf(y0) - 1.0f);
    y1 = (y1 > 0.0f) ? y1 : (expf(y1) - 1.0f);
    float* op = out + (size_t)n * DIM;
    op[lane]      = y0;
    op[lane + 32] = y1;
}

// ---------- output head: d_out[n] = h[n,:] . Wout + bout ----------
__global__ void k_out(const float* __restrict__ hfin, const float* __restrict__ Wout,
                      const float* __restrict__ bout, float* __restrict__ out, int N) {
    int wave = threadIdx.x >> 5;
    int lane = threadIdx.x & 31;
    int n = blockIdx.x * 8 + wave;
    if (n >= N) return;
    const float* hp = hfin + (size_t)n * DIM;
    float p = hp[lane] * Wout[lane] + hp[lane + 32] * Wout[lane + 32];
    #pragma unroll
    for (int m = 16; m >= 1; m >>= 1) p += __shfl_xor(p, m, 32);
    if (lane == 0) out[n] = p + bout[0];
}

extern "C" void kernel_launch(void* const* d_in, const int* in_sizes, int n_in,
                              void* d_out, int out_size, void* d_ws, size_t ws_size,
                              hipStream_t stream) {
    const float* x    = (const float*)d_in[0];
    const int*   ei   = (const int*)  d_in[1];
    const float* eatt = (const float*)d_in[3];
    const float* W1   = (const float*)d_in[4];
    const float* as1  = (const float*)d_in[5];
    const float* ad1  = (const float*)d_in[6];
    const float* We1  = (const float*)d_in[7];
    const float* ae1  = (const float*)d_in[8];
    const float* b1   = (const float*)d_in[9];
    const float* W2   = (const float*)d_in[10];
    const float* as2  = (const float*)d_in[11];
    const float* ad2  = (const float*)d_in[12];
    const float* We2  = (const float*)d_in[13];
    const float* ae2  = (const float*)d_in[14];
    const float* b2   = (const float*)d_in[15];
    const float* g1   = (const float*)d_in[16];
    const float* be1  = (const float*)d_in[17];
    const float* g2   = (const float*)d_in[18];
    const float* be2  = (const float*)d_in[19];
    const float* Wout = (const float*)d_in[20];
    const float* bout = (const float*)d_in[21];
    float* outp = (float*)d_out;

    const int N = in_sizes[0] / DIM;
    const int E = in_sizes[1] / 2;
    const long ET = (long)E + N;

    // workspace layout (floats)
    float* w = (float*)d_ws;
    size_t o = 0;
    float*    lattr = w + o; o += N;            // loop sum -> loop mean
    float*    cnt   = w + o; o += N;
    float*    xs    = w + o; o += (size_t)N * DIM;
    float*    a_src = w + o; o += (size_t)N * HEADS;
    float*    a_dst = w + o; o += (size_t)N * HEADS;
    unsigned* mkey  = (unsigned*)(w + o); o += (size_t)N * HEADS;
    float*    den   = w + o; o += (size_t)N * HEADS;
    float*    agg   = w + o; o += (size_t)N * DIM;
    float*    h1    = w + o; o += (size_t)N * DIM;
    float*    exv   = w + o; o += (size_t)ET * HEADS;
    float*    sed   = w + o; o += 8;

    auto cdiv = [](long a, long b) { return (int)((a + b - 1) / b); };
    const int T = 256;

    // self-loop attrs
    k_zero<<<512, T, 0, stream>>>(lattr, (long)2 * N);
    k_edge_count<<<cdiv(E, T), T, 0, stream>>>(ei, eatt, lattr, cnt, E);
    k_loop_attr<<<cdiv(N, T), T, 0, stream>>>(lattr, cnt, N);
    k_edge_scal<<<1, 32, 0, stream>>>(We1, ae1, We2, ae2, sed);

    // ---------------- layer 1 ----------------
    k_gemm64_wmma<<<cdiv((long)cdiv(N, 16), 2), T, 0, stream>>>(x, W1, xs, N);
    k_attn_coef<<<cdiv((long)N * HEADS, T), T, 0, stream>>>(xs, as1, ad1, a_src, a_dst, N);
    k_zero<<<2048, T, 0, stream>>>((float*)mkey, (long)N * (HEADS + HEADS + DIM)); // mkey|den|agg
    k_edge_max<<<cdiv(ET, T), T, 0, stream>>>(ei, eatt, lattr, sed, a_src, a_dst, mkey, E, N);
    k_edge_exp<<<cdiv(ET, T), T, 0, stream>>>(ei, eatt, lattr, sed, a_src, a_dst, mkey, exv, den, E, N);
    k_edge_agg<<<cdiv(ET * DIM, T), T, 0, stream>>>(ei, exv, den, xs, agg, E, N);
    k_post<<<cdiv(N, 8), T, 0, stream>>>(agg, x, b1, g1, be1, h1, N);

    // ---------------- layer 2 ----------------
    k_gemm64_wmma<<<cdiv((long)cdiv(N, 16), 2), T, 0, stream>>>(h1, W2, xs, N);
    k_attn_coef<<<cdiv((long)N * HEADS, T), T, 0, stream>>>(xs, as2, ad2, a_src, a_dst, N);
    k_zero<<<2048, T, 0, stream>>>((float*)mkey, (long)N * (HEADS + HEADS + DIM));
    k_edge_max<<<cdiv(ET, T), T, 0, stream>>>(ei, eatt, lattr, sed + 4, a_src, a_dst, mkey, E, N);
    k_edge_exp<<<cdiv(ET, T), T, 0, stream>>>(ei, eatt, lattr, sed + 4, a_src, a_dst, mkey, exv, den, E, N);
    k_edge_agg<<<cdiv(ET * DIM, T), T, 0, stream>>>(ei, exv, den, xs, agg, E, N);
    k_post<<<cdiv(N, 8), T, 0, stream>>>(agg, h1, b2, g2, be2, agg, N);  // in-place per wave: safe

    // output head
    k_out<<<cdiv(N, 8), T, 0, stream>>>(agg, Wout, bout, outp, N);
}